// SemanticLayer_66941360276127
// MI455X (gfx1250) — compile-verified
//
#include <hip/hip_runtime.h>
#include <hip/hip_bf16.h>

#define N_ENT  100000
#define N_REL  100
#define N_EDGE 300000
#define H_DIM  128
#define NBASES 30
#define LDSP   132   // LDS row pitch (floats): 132%64=4 -> conflict-free column reads

typedef float v2f __attribute__((ext_vector_type(2)));
typedef float v8f __attribute__((ext_vector_type(8)));

// D = A(16x4,f32) * B(4x16,f32) + C(16x16,f32), fp32 accumulate (reference-exact precision)
#define WMMA_F32(a, b, c) \
    __builtin_amdgcn_wmma_f32_16x16x4_f32(false, (a), false, (b), (short)0, (c), false, false)

// ---- fast transcendentals: use gfx1250 hardware V_TANH_F32 when available --
#if __has_builtin(__builtin_amdgcn_tanhf)
__device__ __forceinline__ float fast_tanh(float x) { return __builtin_amdgcn_tanhf(x); }
__device__ __forceinline__ float fast_sigmoid(float x) {
    return fmaf(0.5f, __builtin_amdgcn_tanhf(0.5f * x), 0.5f);
}
#else
__device__ __forceinline__ float fast_tanh(float x) { return tanhf(x); }
__device__ __forceinline__ float fast_sigmoid(float x) {
    return __builtin_amdgcn_rcpf(1.0f + __expf(-x));   // v_rcp_f32, no IEEE div sequence
}
#endif

// ---------------------------------------------------------------------------
// GRU on a 16-row tile whose input sits in LDS (16 x H_DIM, pitch LDSP).
// 8 waves; wave w owns output columns [16w,16w+16) of each of the 3 gates:
// 3 WMMA chains per wave, gating fully in registers.
// gi = x @ w_ih^T + b_ih ; r=sig(gi_r+b_hh_r); z=sig(gi_z+b_hh_z);
// n=tanh(gi_n + r*b_hh_n); h=(1-z)*n  [optionally tanh(h)]
// ---------------------------------------------------------------------------
__device__ __forceinline__ void gru_from_lds(const float* __restrict__ sA,
                                             const float* __restrict__ w_ihT, // [128][384]
                                             const float* __restrict__ b_ih,
                                             const float* __restrict__ b_hh,
                                             float* __restrict__ out,
                                             int row0, bool do_tanh)
{
    const int lane = threadIdx.x & 31;
    const int w    = threadIdx.x >> 5;
    const int m16  = lane & 15;
    const int kk   = (lane >> 4) << 1;   // lanes 16-31 hold K+2,K+3
    const int c0   = w * 16 + m16;       // column within one gate block

    v8f cr = {}, cz = {}, cn = {};
    for (int k = 0; k < H_DIM; k += 4) {
        v2f a;
        a.x = sA[m16 * LDSP + k + kk];
        a.y = sA[m16 * LDSP + k + kk + 1];
        const float* B0 = w_ihT + (size_t)(k + kk) * 384;
        const float* B1 = B0 + 384;
        v2f br; br.x = B0[c0];       br.y = B1[c0];
        v2f bz; bz.x = B0[128 + c0]; bz.y = B1[128 + c0];
        v2f bn; bn.x = B0[256 + c0]; bn.y = B1[256 + c0];
        cr = WMMA_F32(a, br, cr);
        cz = WMMA_F32(a, bz, cz);
        cn = WMMA_F32(a, bn, cn);
    }

    const int   j   = c0;
    const float bir = b_ih[j], biz = b_ih[128 + j], bin_ = b_ih[256 + j];
    const float hr  = b_hh[j], hz  = b_hh[128 + j], hn   = b_hh[256 + j];
    const int mbase = (lane >> 4) << 3;   // C layout: VGPR v -> M = v (+8 for hi lanes)
#pragma unroll
    for (int v = 0; v < 8; ++v) {
        float ir = cr[v] + bir, iz = cz[v] + biz, in_ = cn[v] + bin_;
        float r  = fast_sigmoid(ir + hr);
        float z  = fast_sigmoid(iz + hz);
        float n  = fast_tanh(in_ + r * hn);
        float hv = (1.0f - z) * n;
        out[(size_t)(row0 + mbase + v) * H_DIM + j] = do_tanh ? fast_tanh(hv) : hv;
    }
}

// ---- GRU #1: h = gru(ent_emb) -> out (d_out used as scratch for h) ---------
__global__ __launch_bounds__(256) void k_gru1(const float* __restrict__ x,
                                              const float* __restrict__ w_ihT,
                                              const float* __restrict__ b_ih,
                                              const float* __restrict__ b_hh,
                                              float* __restrict__ out)
{
    __shared__ float sA[16 * LDSP];
    const int row0 = blockIdx.x * 16;
    for (int idx = threadIdx.x; idx < 16 * 32; idx += 256) {
        int i = idx >> 5, c = (idx & 31) << 2;
        const float4 v = *(const float4*)(x + (size_t)(row0 + i) * H_DIM + c);
        float* p = sA + i * LDSP + c;
        p[0] = v.x; p[1] = v.y; p[2] = v.z; p[3] = v.w;
    }
    __syncthreads();
    gru_from_lds(sA, w_ihT, b_ih, b_hh, out, row0, false);
}

// ---- W_r = sum_m comp[r,m] * bases[m]  (100 x 128 x 128) -------------------
__global__ __launch_bounds__(256) void k_wrel(const float* __restrict__ bases,
                                              const float* __restrict__ comp,
                                              float* __restrict__ W)
{
    int idx = blockIdx.x * 256 + threadIdx.x;       // 100*16384 exactly
    int r = idx >> 14, p = idx & 16383;
    float acc = 0.f;
#pragma unroll
    for (int m = 0; m < NBASES; ++m)
        acc = fmaf(comp[r * NBASES + m], bases[(size_t)m * 16384 + p], acc);
    W[idx] = acc;
}

// ---- transpose w_ih [384][128] -> w_ihT [128][384] for coalesced B loads ---
__global__ __launch_bounds__(256) void k_twih(const float* __restrict__ w_ih,
                                              float* __restrict__ w_ihT)
{
    int idx = blockIdx.x * 256 + threadIdx.x;       // 384*128 exactly
    int j = idx >> 7, k = idx & 127;
    w_ihT[(size_t)k * 384 + j] = w_ih[idx];
}

// ---- per-edge stats: in-degree + relation histogram ------------------------
__global__ __launch_bounds__(256) void k_edge_stats(const int* __restrict__ dst,
                                                    const int* __restrict__ rel,
                                                    unsigned* __restrict__ deg,
                                                    unsigned* __restrict__ cnt, int n)
{
    int e = blockIdx.x * 256 + threadIdx.x;
    if (e < n) {
        atomicAdd(&deg[dst[e]], 1u);
        atomicAdd(&cnt[rel[e]], 1u);
    }
}

__global__ __launch_bounds__(256) void k_norm(const unsigned* __restrict__ deg,
                                              float* __restrict__ nrm, int n)
{
    int i = blockIdx.x * 256 + threadIdx.x;
    if (i < n) {
        unsigned d = deg[i];
        nrm[i] = d ? __builtin_amdgcn_rcpf((float)d) : 0.0f;
    }
}

// ---- 100-bin exclusive scan (tiny; serial) ---------------------------------
__global__ void k_scan(const unsigned* __restrict__ cnt,
                       unsigned* __restrict__ base, unsigned* __restrict__ cursor)
{
    if (threadIdx.x == 0 && blockIdx.x == 0) {
        unsigned run = 0;
        for (int r = 0; r < N_REL; ++r) { base[r] = run; cursor[r] = run; run += cnt[r]; }
    }
}

// ---- bucket edges by relation ----------------------------------------------
__global__ __launch_bounds__(256) void k_scatter(const int* __restrict__ rel,
                                                 unsigned* __restrict__ cursor,
                                                 int* __restrict__ perm, int n)
{
    int e = blockIdx.x * 256 + threadIdx.x;
    if (e < n) perm[atomicAdd(&cursor[rel[e]], 1u)] = e;
}

// ---------------------------------------------------------------------------
// Message pass: per (relation r, 64-edge tile): [64x128] x W_r[128x128] via
// f32 WMMA; scale by norm[dst]; scatter-add into agg. A staged in LDS.
// 8 waves: wave w -> m-tile (w&3), n-tiles 4*(w>>2)..+3  (4 accumulators).
// ---------------------------------------------------------------------------
__global__ __launch_bounds__(256) void k_msg(const float* __restrict__ h,
                                             const float* __restrict__ W,
                                             const int* __restrict__ src,
                                             const int* __restrict__ dst,
                                             const float* __restrict__ nrm,
                                             const int* __restrict__ perm,
                                             const unsigned* __restrict__ cnt,
                                             const unsigned* __restrict__ baseo,
                                             float* __restrict__ agg)
{
    const int r = blockIdx.y;
    const unsigned n_r = cnt[r];
    const unsigned t0  = blockIdx.x * 64u;
    if (t0 >= n_r) return;

    __shared__ float sA[64 * LDSP];
    __shared__ int   s_src[64];
    __shared__ int   s_dst[64];
    __shared__ float s_scale[64];

    const unsigned b0 = baseo[r];
    if (threadIdx.x < 64) {
        int i = threadIdx.x;
        unsigned loc = t0 + (unsigned)i;
        if (loc < n_r) {
            int e = perm[b0 + loc];
            int d = dst[e];
            s_src[i] = src[e]; s_dst[i] = d; s_scale[i] = nrm[d];
        } else {
            s_src[i] = 0; s_dst[i] = -1; s_scale[i] = 0.0f;
        }
    }
    __syncthreads();

    for (int idx = threadIdx.x; idx < 64 * 32; idx += 256) {
        int i = idx >> 5, c = (idx & 31) << 2;
        const float4 v = *(const float4*)(h + (size_t)s_src[i] * H_DIM + c);
        float* p = sA + i * LDSP + c;
        p[0] = v.x; p[1] = v.y; p[2] = v.z; p[3] = v.w;
    }
    __syncthreads();

    const float* Wr  = W + (size_t)r * 16384;
    const int lane = threadIdx.x & 31;
    const int w    = threadIdx.x >> 5;
    const int m16  = lane & 15;
    const int kk   = (lane >> 4) << 1;
    const int mt   = w & 3;
    const int ntg  = (w >> 2) << 2;
    const int arow = mt * 16 + m16;
    const int ncol = ntg * 16 + m16;

    v8f a0 = {}, a1 = {}, a2 = {}, a3 = {};
    for (int k = 0; k < H_DIM; k += 4) {
        v2f a;
        a.x = sA[arow * LDSP + k + kk];
        a.y = sA[arow * LDSP + k + kk + 1];
        const float* B0 = Wr + (size_t)(k + kk) * H_DIM;
        const float* B1 = B0 + H_DIM;
        v2f b;
        b.x = B0[ncol];      b.y = B1[ncol];      a0 = WMMA_F32(a, b, a0);
        b.x = B0[ncol + 16]; b.y = B1[ncol + 16]; a1 = WMMA_F32(a, b, a1);
        b.x = B0[ncol + 32]; b.y = B1[ncol + 32]; a2 = WMMA_F32(a, b, a2);
        b.x = B0[ncol + 48]; b.y = B1[ncol + 48]; a3 = WMMA_F32(a, b, a3);
    }

    const int mbase = mt * 16 + ((lane >> 4) << 3);
#pragma unroll
    for (int v = 0; v < 8; ++v) {
        int mi = mbase + v;
        int d  = s_dst[mi];
        if (d >= 0) {
            float sc = s_scale[mi];
            float* ap = agg + (size_t)d * H_DIM;
            atomicAdd(ap + ncol,      a0[v] * sc);
            atomicAdd(ap + ncol + 16, a1[v] * sc);
            atomicAdd(ap + ncol + 32, a2[v] * sc);
            atomicAdd(ap + ncol + 48, a3[v] * sc);
        }
    }
}

// ---- fused: h2 = agg + h@loop_weight ; out = tanh(gru(h2)) -----------------
__global__ __launch_bounds__(256) void k_gru2(const float* __restrict__ h,
                                              const float* __restrict__ agg,
                                              const float* __restrict__ lw,
                                              const float* __restrict__ w_ihT,
                                              const float* __restrict__ b_ih,
                                              const float* __restrict__ b_hh,
                                              float* __restrict__ out)
{
    __shared__ float sA[16 * LDSP];
    __shared__ float sH[16 * LDSP];
    const int row0 = blockIdx.x * 16;
    for (int idx = threadIdx.x; idx < 16 * 32; idx += 256) {
        int i = idx >> 5, c = (idx & 31) << 2;
        const float4 v = *(const float4*)(h + (size_t)(row0 + i) * H_DIM + c);
        float* p = sA + i * LDSP + c;
        p[0] = v.x; p[1] = v.y; p[2] = v.z; p[3] = v.w;
    }
    __syncthreads();

    const int lane = threadIdx.x & 31;
    const int w    = threadIdx.x >> 5;
    const int m16  = lane & 15;
    const int kk   = (lane >> 4) << 1;
    const int n    = w * 16 + m16;

    v8f acc = {};
    for (int k = 0; k < H_DIM; k += 4) {
        v2f a;
        a.x = sA[m16 * LDSP + k + kk];
        a.y = sA[m16 * LDSP + k + kk + 1];
        v2f b;
        b.x = lw[(size_t)(k + kk) * H_DIM + n];
        b.y = lw[(size_t)(k + kk + 1) * H_DIM + n];
        acc = WMMA_F32(a, b, acc);
    }
    const int mbase = (lane >> 4) << 3;
#pragma unroll
    for (int v = 0; v < 8; ++v) {
        int m = mbase + v;
        sH[m * LDSP + n] = acc[v] + agg[(size_t)(row0 + m) * H_DIM + n];
    }
    __syncthreads();
    gru_from_lds(sH, w_ihT, b_ih, b_hh, out, row0, true);
}

// ---------------------------------------------------------------------------
extern "C" void kernel_launch(void* const* d_in, const int* in_sizes, int n_in,
                              void* d_out, int out_size, void* d_ws, size_t ws_size,
                              hipStream_t stream)
{
    (void)in_sizes; (void)n_in; (void)out_size; (void)ws_size;
    const float* ent   = (const float*)d_in[0];
    // d_in[1] rel_emb: unused by the reference
    const int*   src   = (const int*)d_in[2];
    const int*   dst   = (const int*)d_in[3];
    const int*   rel   = (const int*)d_in[4];
    const float* bases = (const float*)d_in[5];
    const float* comp  = (const float*)d_in[6];
    const float* lw    = (const float*)d_in[7];
    const float* w_ih  = (const float*)d_in[8];
    const float* b_ih  = (const float*)d_in[9];
    const float* b_hh  = (const float*)d_in[10];
    float* out = (float*)d_out;

    char* ws = (char*)d_ws;
    size_t off = 0;
    auto alloc = [&](size_t bytes) -> void* {
        void* p = ws + off;
        off += (bytes + 255) & ~(size_t)255;
        return p;
    };
    float*    agg    = (float*)alloc((size_t)N_ENT * H_DIM * 4);     // 51.2 MB
    float*    W      = (float*)alloc((size_t)N_REL * 16384 * 4);     //  6.6 MB
    float*    w_ihT  = (float*)alloc((size_t)384 * 128 * 4);
    float*    nrm    = (float*)alloc((size_t)N_ENT * 4);
    unsigned* deg    = (unsigned*)alloc((size_t)N_ENT * 4);
    unsigned* cnt    = (unsigned*)alloc(128 * 4);
    unsigned* baseo  = (unsigned*)alloc(128 * 4);
    unsigned* cursor = (unsigned*)alloc(128 * 4);
    int*      perm   = (int*)alloc((size_t)N_EDGE * 4);

    hipMemsetAsync(agg, 0, (size_t)N_ENT * H_DIM * 4, stream);
    hipMemsetAsync(deg, 0, (size_t)N_ENT * 4, stream);
    hipMemsetAsync(cnt, 0, 128 * 4, stream);

    k_wrel<<<6400, 256, 0, stream>>>(bases, comp, W);
    k_twih<<<192, 256, 0, stream>>>(w_ih, w_ihT);
    k_gru1<<<N_ENT / 16, 256, 0, stream>>>(ent, w_ihT, b_ih, b_hh, out);   // h -> out
    k_edge_stats<<<(N_EDGE + 255) / 256, 256, 0, stream>>>(dst, rel, deg, cnt, N_EDGE);
    k_norm<<<(N_ENT + 255) / 256, 256, 0, stream>>>(deg, nrm, N_ENT);
    k_scan<<<1, 32, 0, stream>>>(cnt, baseo, cursor);
    k_scatter<<<(N_EDGE + 255) / 256, 256, 0, stream>>>(rel, cursor, perm, N_EDGE);
    k_msg<<<dim3(128, N_REL), 256, 0, stream>>>(out, W, src, dst, nrm, perm, cnt, baseo, agg);
    k_gru2<<<N_ENT / 16, 256, 0, stream>>>(out, agg, lw, w_ihT, b_ih, b_hh, out);
}